// OptLayer_9749575762688
// MI455X (gfx1250) — compile-verified
//
#include <hip/hip_runtime.h>
#include <stdint.h>

// Projection of each row y (N=1024) onto {z : 0 <= z <= upper, sum z = 512}
// via active-set water-filling, matching the JAX reference semantics.
//
// Design (MI455X / gfx1250, wave32):
//  - one wave per row; each lane owns 32 elements (8 x float4), fully register
//    resident across all iterations -> single pass over HBM (~16.8 MB total,
//    ~0.7 us at 23.3 TB/s; compute is negligible).
//  - `upper` staged global->LDS once per workgroup with the CDNA5 async path
//    (global_load_async_to_lds_b128 + s_wait_asynccnt), then ds_load_b128.
//  - per-iteration reductions are incremental deltas (newly clamped elements
//    only), reduced intra-wave with 5 shuffle steps; wave-uniform early exit
//    when phase==2 (identical to the reference's no-op tail iterations).

constexpr int   N_COLS          = 1024;
constexpr int   LANES           = 32;                    // wave32
constexpr int   VECS            = N_COLS / (LANES * 4);  // 8 float4 per lane
constexpr int   WAVES_PER_BLOCK = 8;
constexpr int   BLOCK           = LANES * WAVES_PER_BLOCK; // 256
constexpr float NBIKES          = 512.0f;
constexpr int   N_ITERS         = 32;

__global__ __launch_bounds__(BLOCK) void waterfill_project_kernel(
    const float* __restrict__ y,
    const float* __restrict__ upper,
    float* __restrict__ out,
    int nrows)
{
    __shared__ float sh_upper[N_COLS];

    const int tid = threadIdx.x;

    // ---- Stage `upper` (4 KB) into LDS via async global->LDS copy ----
    // 256 threads x 16 B each = 4096 B. One vector instruction per wave;
    // tracked by ASYNCcnt, then made visible WG-wide by the barrier.
    {
        uint32_t lds_off = (uint32_t)(uintptr_t)(&sh_upper[tid * 4]);
        uint64_t gaddr   = (uint64_t)(uintptr_t)(upper + tid * 4);
        asm volatile("global_load_async_to_lds_b128 %0, %1, off"
                     :
                     : "v"(lds_off), "v"(gaddr)
                     : "memory");
        asm volatile("s_wait_asynccnt 0" ::: "memory");
    }
    __syncthreads();

    const int lane = tid & (LANES - 1);
    const int wave = tid >> 5;
    const int row  = blockIdx.x * WAVES_PER_BLOCK + wave;
    if (row >= nrows) return;

    // ---- Load row + upper into registers (coalesced b128) ----
    float4 yv[VECS], uv[VECS];
    const float4* __restrict__ yrow = (const float4*)(y + (size_t)row * N_COLS);
    const float4* __restrict__ shv  = (const float4*)sh_upper;
#pragma unroll
    for (int k = 0; k < VECS; ++k) {
        yv[k] = yrow[lane + LANES * k];   // global_load_b128
        uv[k] = shv[lane + LANES * k];    // ds_load_b128
    }

    // ---- Wave-uniform running state ----
    uint32_t alive   = 0xFFFFFFFFu;  // bit (4k+c): element unclamped
    uint32_t atUpper = 0u;           // clamped-at-upper marker

    float S = 0.0f;                  // sum of y over unclamped set
#pragma unroll
    for (int k = 0; k < VECS; ++k)
        S += yv[k].x + yv[k].y + yv[k].z + yv[k].w;
#pragma unroll
    for (int off = 16; off > 0; off >>= 1)
        S += __shfl_xor(S, off, 32);

    float C     = NBIKES;
    int   n_un  = N_COLS;
    int   phase = 0;

    // ---- Active-set iterations (wave-uniform loop, early exit at phase 2) ----
    for (int it = 0; it < N_ITERS && phase < 2; ++it) {
        const float add = (C - S) / fmaxf((float)n_un, 1.0f);

        float dS = 0.0f, dC = 0.0f;
        int   dn = 0;
#pragma unroll
        for (int k = 0; k < VECS; ++k) {
            const float ya[4] = {yv[k].x, yv[k].y, yv[k].z, yv[k].w};
            const float ua[4] = {uv[k].x, uv[k].y, uv[k].z, uv[k].w};
#pragma unroll
            for (int c = 0; c < 4; ++c) {
                const uint32_t bit = 1u << (4 * k + c);
                if (alive & bit) {
                    const float z    = ya[c] + add;
                    const bool  viol = (phase == 0) ? (z < 0.0f) : (z > ua[c]);
                    if (viol) {
                        alive &= ~bit;
                        dn    += 1;
                        dS    += ya[c];
                        if (phase == 1) { dC += ua[c]; atUpper |= bit; }
                    }
                }
            }
        }

        // 3 wave reductions of the deltas (5 shuffle steps each)
#pragma unroll
        for (int off = 16; off > 0; off >>= 1) {
            dS += __shfl_xor(dS, off, 32);
            dC += __shfl_xor(dC, off, 32);
            dn += __shfl_xor(dn, off, 32);
        }

        S    -= dS;
        C    -= dC;     // phase 0 clamps to lower=0, so dC==0 there
        n_un -= dn;
        if (dn == 0) phase++;   // no violators this round -> advance phase
    }

    // ---- Final water-fill + store ----
    const float add = (C - S) / fmaxf((float)n_un, 1.0f);
    float4* __restrict__ orow = (float4*)(out + (size_t)row * N_COLS);
#pragma unroll
    for (int k = 0; k < VECS; ++k) {
        const float ya[4] = {yv[k].x, yv[k].y, yv[k].z, yv[k].w};
        const float ua[4] = {uv[k].x, uv[k].y, uv[k].z, uv[k].w};
        float oa[4];
#pragma unroll
        for (int c = 0; c < 4; ++c) {
            const uint32_t bit = 1u << (4 * k + c);
            oa[c] = (alive & bit) ? (ya[c] + add)
                                  : ((atUpper & bit) ? ua[c] : 0.0f);
        }
        orow[lane + LANES * k] = make_float4(oa[0], oa[1], oa[2], oa[3]);
    }
}

extern "C" void kernel_launch(void* const* d_in, const int* in_sizes, int n_in,
                              void* d_out, int out_size, void* d_ws, size_t ws_size,
                              hipStream_t stream) {
    (void)n_in; (void)out_size; (void)d_ws; (void)ws_size;

    const float* y     = (const float*)d_in[0];
    const float* upper = (const float*)d_in[1];
    float*       out   = (float*)d_out;

    const int ncols = in_sizes[1];            // 1024 (kernel specialized for this)
    const int nrows = in_sizes[0] / ncols;    // 2048

    const int grid = (nrows + WAVES_PER_BLOCK - 1) / WAVES_PER_BLOCK;
    waterfill_project_kernel<<<grid, BLOCK, 0, stream>>>(y, upper, out, nrows);
}